// GNNEncoder_58428735095225
// MI455X (gfx1250) — compile-verified
//
#include <hip/hip_runtime.h>
#include <hip/hip_bf16.h>

// ---------------------------------------------------------------------------
// GCN encoder for MI455X (gfx1250).
// Dense transforms use V_WMMA_F32_16X16X4_F32 (exact f32 matrix math — the
// GEMMs are ~1.2 GFLOP total, far below any compute ceiling, so we keep full
// precision). The edge gather/scatter phase is the real cost (~1 GB of L2
// traffic, which fits the 192 MB global L2) and is done with coalesced
// 64-wide rows + f32 L2 atomics.
// ---------------------------------------------------------------------------

typedef __attribute__((ext_vector_type(2))) float v2f;
typedef __attribute__((ext_vector_type(8))) float v8f;

#define N_NODES    50000
#define N_EDGES    800000
#define IN_DIM     128
#define HID_DIM    64
#define OUT_DIM    64
#define NUM_GRAPHS 512

// -------------------- degree / normalization --------------------

__global__ void degree_kernel(const int* __restrict__ dst, float* __restrict__ deg) {
    int e = blockIdx.x * 256 + threadIdx.x;
    if (e < N_EDGES) atomicAdd(deg + dst[e], 1.0f);
}

__global__ void dinv_kernel(float* __restrict__ deg) {
    int n = blockIdx.x * 256 + threadIdx.x;
    if (n < N_NODES) deg[n] = rsqrtf(fmaxf(deg[n], 1.0f));
}

// -------------------- WMMA GEMM: H = (relu?)(A) @ W + b --------------------
// One 128-thread block (4 waves) handles a 16-row node tile; wave w computes
// output columns [16w, 16w+16) with a chain of v_wmma_f32_16x16x4_f32.
//
// f32 WMMA VGPR layout (ISA 7.12.2):
//   A 16x4 : lane%16 = M row; VGPR0/1 hold K = 2*(lane/16) + {0,1}
//   B 4x16 : lane%16 = N col; VGPR0/1 hold K = 2*(lane/16) + {0,1}
//   C/D    : 8 VGPRs, element v maps to row M = v + 8*(lane/16), col = lane%16

template <int K, bool RELU_IN>
__global__ void gcn_gemm_wmma(const float* __restrict__ A,  // [N_NODES, K]
                              const float* __restrict__ W,  // [K, 64]
                              const float* __restrict__ B,  // [64]
                              float* __restrict__ H)        // [N_NODES, 64]
{
    const int wave  = threadIdx.x >> 5;   // 0..3 -> 16-col tile
    const int lane  = threadIdx.x & 31;
    const int half  = lane >> 4;          // 0/1
    const int l16   = lane & 15;
    const int m0    = blockIdx.x * 16;
    const int n0    = wave * 16;

    // C starts as broadcast bias (depends only on column).
    const float bias = B[n0 + l16];
    v8f acc;
#pragma unroll
    for (int i = 0; i < 8; ++i) acc[i] = bias;

    const float* arow = A + (size_t)(m0 + l16) * K;

#pragma unroll 4
    for (int k0 = 0; k0 < K; k0 += 4) {
        const int ka = k0 + 2 * half;
        v2f a;
        if (RELU_IN) {                       // fuse previous layer's ReLU
            a.x = fmaxf(arow[ka], 0.0f);
            a.y = fmaxf(arow[ka + 1], 0.0f);
        } else {
            a = *(const v2f*)(arow + ka);    // 8B-aligned float2 load
        }
        v2f b;
        b.x = W[(size_t)ka * 64 + n0 + l16];
        b.y = W[(size_t)(ka + 1) * 64 + n0 + l16];

        // (neg_a, A, neg_b, B, c_mod, C, reuse_a, reuse_b)
        acc = __builtin_amdgcn_wmma_f32_16x16x4_f32(
            false, a, false, b, (short)0, acc, false, false);
    }

    float* hrow = H + (size_t)m0 * 64 + n0 + l16;
#pragma unroll
    for (int v = 0; v < 8; ++v)
        hrow[(size_t)(v + 8 * half) * 64] = acc[v];
}

// -------------------- edge scatter: agg[dst] += h[src] * norm --------------
// 256 threads = 4 edges x 64 dims; consecutive lanes cover consecutive dims
// of one source row -> fully coalesced gather + segment-coalesced atomics.

__global__ void gcn_scatter(const float* __restrict__ H,
                            const int*  __restrict__ src,
                            const int*  __restrict__ dst,
                            const float* __restrict__ dinv,
                            float* __restrict__ AGG)
{
    const int e = blockIdx.x * 4 + (threadIdx.x >> 6);
    const int d = threadIdx.x & 63;
    if (e >= N_EDGES) return;
    const int s = src[e];
    const int t = dst[e];
    const float nrm = dinv[s] * dinv[t];
    atomicAdd(AGG + (size_t)t * 64 + d, H[(size_t)s * 64 + d] * nrm);
}

// -------------------- pooling --------------------

__global__ void pool_cnt(const int* __restrict__ batch, float* __restrict__ cnt) {
    int n = blockIdx.x * 256 + threadIdx.x;
    if (n < N_NODES) atomicAdd(cnt + batch[n], 1.0f);
}

__global__ void pool_sum(const float* __restrict__ AGG,   // agg2 (pre-ReLU)
                         const int*  __restrict__ batch,
                         float* __restrict__ sums)
{
    const int n = blockIdx.x * 4 + (threadIdx.x >> 6);
    const int d = threadIdx.x & 63;
    if (n >= N_NODES) return;
    const float v = fmaxf(AGG[(size_t)n * 64 + d], 0.0f);  // fused final ReLU
    atomicAdd(sums + (size_t)batch[n] * 64 + d, v);
}

__global__ void pool_div(const float* __restrict__ sums,
                         const float* __restrict__ cnt,
                         float* __restrict__ out)
{
    int i = blockIdx.x * 256 + threadIdx.x;
    if (i < NUM_GRAPHS * 64)
        out[i] = sums[i] / fmaxf(cnt[i >> 6], 1.0f);
}

// -------------------- launch --------------------

extern "C" void kernel_launch(void* const* d_in, const int* in_sizes, int n_in,
                              void* d_out, int out_size, void* d_ws, size_t ws_size,
                              hipStream_t stream)
{
    const float* x  = (const float*)d_in[0];
    const int*   ei = (const int*)  d_in[1];   // [2, N_EDGES]
    const int*   bt = (const int*)  d_in[2];   // [N_NODES]
    const float* W1 = (const float*)d_in[3];
    const float* b1 = (const float*)d_in[4];
    const float* W2 = (const float*)d_in[5];
    const float* b2 = (const float*)d_in[6];
    float* out = (float*)d_out;

    const int* src = ei;
    const int* dst = ei + N_EDGES;

    // workspace layout (floats); buf0 is reused: h1 then agg2
    const size_t NM = (size_t)N_NODES * 64;
    float* buf0 = (float*)d_ws;                   // h1  -> agg2
    float* buf1 = buf0 + NM;                      // agg1
    float* buf2 = buf1 + NM;                      // h2
    float* dinv = buf2 + NM;                      // deg -> deg^{-1/2}
    float* sums = dinv + N_NODES;                 // [512, 64]
    float* cnt  = sums + (size_t)NUM_GRAPHS * 64; // [512]

    // zero-init accumulators (hipMemsetAsync is graph-capture legal)
    hipMemsetAsync(dinv, 0, N_NODES * sizeof(float), stream);
    hipMemsetAsync(buf1, 0, NM * sizeof(float), stream);
    hipMemsetAsync(sums, 0, (size_t)(NUM_GRAPHS * 64 + NUM_GRAPHS) * sizeof(float), stream);

    // symmetric degree normalization
    degree_kernel<<<(N_EDGES + 255) / 256, 256, 0, stream>>>(dst, dinv);
    dinv_kernel<<<(N_NODES + 255) / 256, 256, 0, stream>>>(dinv);

    // layer 1: h1 = x @ W1 + b1 ; agg1 = scatter(norm * h1[src] -> dst)
    gcn_gemm_wmma<IN_DIM, false><<<N_NODES / 16, 128, 0, stream>>>(x, W1, b1, buf0);
    gcn_scatter<<<(N_EDGES + 3) / 4, 256, 0, stream>>>(buf0, src, dst, dinv, buf1);

    // layer 2: h2 = relu(agg1) @ W2 + b2 ; agg2 = scatter(...)
    gcn_gemm_wmma<HID_DIM, true><<<N_NODES / 16, 128, 0, stream>>>(buf1, W2, b2, buf2);
    hipMemsetAsync(buf0, 0, NM * sizeof(float), stream);      // buf0 now agg2
    gcn_scatter<<<(N_EDGES + 3) / 4, 256, 0, stream>>>(buf2, src, dst, dinv, buf0);

    // scatter-mean pooling (final ReLU fused into pool_sum read)
    pool_cnt<<<(N_NODES + 255) / 256, 256, 0, stream>>>(bt, cnt);
    pool_sum<<<(N_NODES + 3) / 4, 256, 0, stream>>>(buf0, bt, sums);
    pool_div<<<(NUM_GRAPHS * 64 + 255) / 256, 256, 0, stream>>>(sums, cnt, out);
}